// DTW_20521353740776
// MI455X (gfx1250) — compile-verified
//
#include <hip/hip_runtime.h>
#include <hip/hip_bf16.h>

typedef __attribute__((ext_vector_type(2))) float v2f;
typedef __attribute__((ext_vector_type(8))) float v8f;

#define BATCH 64
#define NN 512
#define MM 512
#define DIMK 8
#define RSTRIDE 514            // N+2
#define RPLANE (514 * 514)
#define BAND 51                // |i-j| <= 51  (bw=0.1, 511*0.1=51.1)
#define KW 103                 // band width = 2*BAND+1
#define DSTRIDE 106            // padded row stride for banded D (f16)
#define NDIAG (NN + MM - 1)    // 1023

__device__ __forceinline__ float devinf() { return __builtin_inff(); }

// ---------------------------------------------------------------------------
// Kernel 1: initialize Rfull to +inf, R[b,0,0] = 0. (loss written by kernel 2)
// Bandwidth-bound: ~67 MB @ 23.3 TB/s ~= 3 us.
// ---------------------------------------------------------------------------
__global__ void dtw_init_kernel(float* __restrict__ out) {
    const long total = (long)BATCH * RPLANE;
    long idx = (long)blockIdx.x * blockDim.x + threadIdx.x;
    const long stride = (long)gridDim.x * blockDim.x;
    for (; idx < total; idx += stride) {
        long rem = idx % RPLANE;                 // position within one (514,514)
        out[BATCH + idx] = (rem == 0) ? 0.0f : devinf();
    }
}

// ---------------------------------------------------------------------------
// Kernel 2: one block per batch. 256 threads = 8 wave32 waves.
//   Phase 0: row norms x2[i], y2[j] -> LDS
//   Phase 1: banded D = x2 + y2 - 2*X.Y^T via V_WMMA_F32_16X16X4_F32 -> LDS f16
//   Phase 2: banded anti-diagonal DTW recurrence, rows in LDS, 1 barrier/diag
// ---------------------------------------------------------------------------
__global__ void __launch_bounds__(256)
dtw_main_kernel(const float* __restrict__ X, const float* __restrict__ Y,
                float* __restrict__ out) {
    __shared__ _Float16 Dband[NN * DSTRIDE];   // ~106 KB banded distance matrix
    __shared__ float x2s[NN];
    __shared__ float y2s[MM];
    __shared__ float Rrow[3][NN];              // rotating recurrence rows

    const int b   = blockIdx.x;
    const int tid = threadIdx.x;
    const float* Xb = X + (long)b * NN * DIMK;
    const float* Yb = Y + (long)b * MM * DIMK;

    // ---------------- Phase 0: row norms ----------------
    for (int i = tid; i < NN; i += 256) {
        const float* xr = Xb + i * DIMK;
        const float* yr = Yb + i * DIMK;
        float sx = 0.0f, sy = 0.0f;
#pragma unroll
        for (int k = 0; k < DIMK; ++k) {
            sx += xr[k] * xr[k];
            sy += yr[k] * yr[k];
        }
        x2s[i] = sx;
        y2s[i] = sy;
        Rrow[0][i] = devinf();
        Rrow[1][i] = devinf();
        Rrow[2][i] = devinf();
    }
    __syncthreads();

    // ---------------- Phase 1: banded D via WMMA ----------------
    // Tiles: 32 row-tiles (I), 9 col-tiles each (J = I-4 .. I+4) cover |i-j|<=51.
    // A = -2 * X tile (16x8), B = Y^T tile (8x16), two K=4 WMMAs accumulate.
    const int wave = tid >> 5;          // 0..7, uniform per wave
    const int lane = tid & 31;
    const int half = lane >> 4;         // 0 or 1
    const int l16  = lane & 15;
    const int kb   = half * 2;          // K pair selected by lane half (A/B layout)

    for (int tile = wave; tile < 32 * 9; tile += 8) {   // uniform trip count/wave
        const int I = tile / 9;
        const int J = I - 4 + (tile % 9);
        if (J < 0 || J > 31) continue;                  // wave-uniform: EXEC stays full

        const float* xr = Xb + (16 * I + l16) * DIMK;   // A row m = l16
        const float* yr = Yb + (16 * J + l16) * DIMK;   // B col n = l16

        v2f a0, a1, b0, b1;
        a0.x = -2.0f * xr[kb];     a0.y = -2.0f * xr[kb + 1];
        a1.x = -2.0f * xr[kb + 4]; a1.y = -2.0f * xr[kb + 5];
        b0.x = yr[kb];             b0.y = yr[kb + 1];
        b1.x = yr[kb + 4];         b1.y = yr[kb + 5];

        v8f c = {};
        c = __builtin_amdgcn_wmma_f32_16x16x4_f32(false, a0, false, b0,
                                                  (short)0, c, false, false);
        c = __builtin_amdgcn_wmma_f32_16x16x4_f32(false, a1, false, b1,
                                                  (short)0, c, false, false);

        // C layout: VGPR r -> M = r + 8*half, N = l16
#pragma unroll
        for (int r = 0; r < 8; ++r) {
            const int ii = 16 * I + r + 8 * half;
            const int jj = 16 * J + l16;
            const int k  = jj - ii + BAND;
            if ((unsigned)k < (unsigned)KW) {
                float val = c[r] + x2s[ii] + y2s[jj];
                Dband[ii * DSTRIDE + k] = (_Float16)val;
            }
        }
    }
    __syncthreads();

    // ---------------- Phase 2: anti-diagonal DTW ----------------
    float* Rb = out + BATCH + (long)b * RPLANE;

    for (int d = 0; d < NDIAG; ++d) {
        const int cur = d % 3;
        const int p1  = (d + 2) % 3;   // diagonal d-1
        const int p2  = (d + 1) % 3;   // diagonal d-2
#pragma unroll
        for (int s = 0; s < 2; ++s) {
            const int i = tid + s * 256;
            const int j = d - i;
            const int k = j - i + BAND;
            const bool inside = (j >= 0) && (j < MM);
            const bool valid  = inside && ((unsigned)k < (unsigned)KW);
            float r = devinf();
            if (valid) {
                const float Dd = (float)Dband[i * DSTRIDE + k];
                const float r0 = (i == 0) ? ((d == 0) ? 0.0f : devinf())
                                          : Rrow[p2][i - 1];
                const float r1 = (i == 0) ? devinf() : Rrow[p1][i - 1];
                const float r2 = Rrow[p1][i];
                r = Dd + fminf(fminf(r0, r1), r2);
                Rb[(long)(i + 1) * RSTRIDE + (j + 1)] = r;
                if (d == NDIAG - 1 && i == NN - 1) out[b] = r;  // loss
            }
            Rrow[cur][i] = r;
        }
        __syncthreads();
    }
}

extern "C" void kernel_launch(void* const* d_in, const int* in_sizes, int n_in,
                              void* d_out, int out_size, void* d_ws, size_t ws_size,
                              hipStream_t stream) {
    const float* X = (const float*)d_in[0];   // (64, 512, 8) f32
    const float* Y = (const float*)d_in[1];   // (64, 512, 8) f32
    float* out = (float*)d_out;               // [loss(64) | Rfull(64,514,514)]

    dtw_init_kernel<<<1024, 256, 0, stream>>>(out);
    dtw_main_kernel<<<BATCH, 256, 0, stream>>>(X, Y, out);
}